// ActorNetwork_84628035601149
// MI455X (gfx1250) — compile-verified
//
#include <hip/hip_runtime.h>

typedef float v2f __attribute__((ext_vector_type(2)));
typedef float v8f __attribute__((ext_vector_type(8)));

// Problem constants (match reference).
constexpr int NNODES = 50000;
constexpr int OBS    = 128;
constexpr int F1     = 80;
constexpr int F2     = 40;
constexpr int AC     = 32;
constexpr float EPS_BN = 1e-5f;
constexpr float SLOPE  = 0.2f;

// Workspace layout (float offsets). All offsets multiple of 4 -> 16B aligned.
constexpr long OFF_AGGR = 0;                       // N*F1 = 4,000,000
constexpr long OFF_M    = 4000000;                 // 50,000 (uint, encoded max)
constexpr long OFF_DEN  = 4050000;                 // 50,000
constexpr long OFF_SUM  = 4100000;                 // 400  (s1,q1,s2,q2,s3,q3)
constexpr long ZEROLEN  = 4100400;                 // zeroed each call
constexpr long OFF_H    = 4100400;                 // N*F1 (h, reused as h2)
constexpr long OFF_H3   = 8100400;                 // N*F2 = 2,000,000
constexpr long OFF_E    = 10100400;                // E+N  = 850,000

__device__ __forceinline__ unsigned enc_ord(float f) {
    unsigned u = __float_as_uint(f);
    return (u & 0x80000000u) ? ~u : (u | 0x80000000u);
}
__device__ __forceinline__ float dec_ord(unsigned u) {
    return __uint_as_float((u & 0x80000000u) ? (u & 0x7fffffffu) : ~u);
}

__global__ void zero_kernel(float* p, long n) {
    long i = (long)blockIdx.x * blockDim.x + threadIdx.x;
    if (i < n) p[i] = 0.0f;
}

// C[M,N] = act(A[M,K] @ B[K,N]) using V_WMMA_F32_16X16X4_F32.
// One wave per 16x16 tile. act: 0=none, 1=relu, 2=sigmoid(relu(x)).
// Out-of-range columns (N=40 case) clamp their B column: junk only reaches
// D columns that are never stored, so the K loop has no divergent branches.
__global__ void gemm_wmma(const float* __restrict__ A, const float* __restrict__ B,
                          float* __restrict__ C, int M, int K, int N,
                          int tilesN, int act) {
    int wave = blockIdx.x * (blockDim.x >> 5) + (threadIdx.x >> 5);
    int lane = threadIdx.x & 31;
    int mt = wave / tilesN;
    int nt = wave % tilesN;
    if (mt * 16 >= M) return;                 // wave-uniform exit: EXEC stays full
    int half = lane >> 4;                     // 0 for lanes 0-15, 1 for 16-31
    int l    = lane & 15;
    int arow = mt * 16 + l;                   // A: lanes hold M=l, K split by half
    int col  = nt * 16 + l;                   // B/C column
    bool colOK = col < N;
    int  bcol  = colOK ? col : (N - 1);       // clamped, always in-bounds

    const float* ap = A + (long)arow * K + 2 * half;   // 8B aligned (K % 4 == 0)
    const float* bp = B + (long)(2 * half) * N + bcol;

    v8f c = {};
    for (int k0 = 0; k0 < K; k0 += 4) {
        v2f a = *(const v2f*)ap;              // A K=k0+2h, k0+2h+1 (b64 load)
        v2f b;
        b.x = bp[0];                          // B row k0+2h
        b.y = bp[N];                          // B row k0+2h+1
        c = __builtin_amdgcn_wmma_f32_16x16x4_f32(false, a, false, b,
                                                  (short)0, c, false, false);
        ap += 4;
        bp += 4 * N;
    }
    if (colOK) {
#pragma unroll
        for (int r = 0; r < 8; ++r) {         // C: VGPR r -> M = r (+8 high half)
            int orow = mt * 16 + r + 8 * half;
            float v = c[r];
            if (act == 1)      v = fmaxf(v, 0.0f);
            else if (act == 2) v = 1.0f / (1.0f + __expf(-fmaxf(v, 0.0f)));
            C[(long)orow * N + col] = v;
        }
    }
}

// One thread per edge: e = att . leaky_relu(h[src]+h[dst]); segment max into mmax.
__global__ void edge_score_kernel(const float* __restrict__ h, const int* __restrict__ ei,
                                  const float* __restrict__ att, float* __restrict__ e,
                                  unsigned* __restrict__ mmax, int E, int total) {
    int i = blockIdx.x * blockDim.x + threadIdx.x;
    if (i >= total) return;
    int s = (i < E) ? ei[i]     : (i - E);
    int d = (i < E) ? ei[E + i] : (i - E);
    const float* hs = h + (long)s * F1;
    const float* hd = h + (long)d * F1;
    float acc = 0.0f;
#pragma unroll
    for (int f = 0; f < F1; f += 4) {
        float4 a = *(const float4*)(att + f);
        float4 xs = *(const float4*)(hs + f);
        float4 xd = *(const float4*)(hd + f);
        float v;
        v = xs.x + xd.x; v = v > 0.0f ? v : SLOPE * v; acc += a.x * v;
        v = xs.y + xd.y; v = v > 0.0f ? v : SLOPE * v; acc += a.y * v;
        v = xs.z + xd.z; v = v > 0.0f ? v : SLOPE * v; acc += a.z * v;
        v = xs.w + xd.w; v = v > 0.0f ? v : SLOPE * v; acc += a.w * v;
    }
    e[i] = acc;
    atomicMax(&mmax[d], enc_ord(acc));
}

// ex = exp(e - max[dst]); denom[dst] += ex (overwrites e with ex).
__global__ void edge_exp_kernel(float* __restrict__ e, const int* __restrict__ ei,
                                const unsigned* __restrict__ mmax,
                                float* __restrict__ denom, int E, int total) {
    int i = blockIdx.x * blockDim.x + threadIdx.x;
    if (i >= total) return;
    int d = (i < E) ? ei[E + i] : (i - E);
    float ex = __expf(e[i] - dec_ord(mmax[d]));
    e[i] = ex;
    atomicAdd(&denom[d], ex);
}

// Wave per edge: aggr[dst] += alpha * h[src] (lane-contiguous float atomics).
__global__ void edge_aggr_kernel(const float* __restrict__ h, const int* __restrict__ ei,
                                 const float* __restrict__ e, const float* __restrict__ denom,
                                 float* __restrict__ aggr, int E, int total) {
    int wv = blockIdx.x * (blockDim.x >> 5) + (threadIdx.x >> 5);
    if (wv >= total) return;
    int lane = threadIdx.x & 31;
    int s = (wv < E) ? ei[wv]     : (wv - E);
    int d = (wv < E) ? ei[E + wv] : (wv - E);
    float alpha = e[wv] / (denom[d] + 1e-16f);
    const float* hs = h + (long)s * F1;
    float* out = aggr + (long)d * F1;
#pragma unroll
    for (int f = lane; f < F1; f += 32)
        atomicAdd(&out[f], alpha * hs[f]);
}

__global__ void bias_add_kernel(float* __restrict__ h, const float* __restrict__ bias,
                                long n, int C) {
    long i = (long)blockIdx.x * blockDim.x + threadIdx.x;
    if (i < n) h[i] += bias[i % C];
}

// Column sums / sum-of-squares: one thread per column, 128-row chunks per block.
__global__ void colstats_kernel(const float* __restrict__ in, int rows, int cols,
                                float* __restrict__ sum, float* __restrict__ sumsq) {
    int c = threadIdx.x;
    if (c >= cols) return;
    int r0 = blockIdx.x * 128;
    int r1 = min(r0 + 128, rows);
    float s = 0.0f, q = 0.0f;
    for (int r = r0; r < r1; ++r) {
        float v = in[(long)r * cols + c];
        s += v; q += v * v;
    }
    atomicAdd(&sum[c], s);
    atomicAdd(&sumsq[c], q);
}

// In-place training-mode batchnorm.
__global__ void bn_kernel(float* __restrict__ h, const float* __restrict__ g,
                          const float* __restrict__ b, const float* __restrict__ sum,
                          const float* __restrict__ sumsq, long n, int C, float invN) {
    long i = (long)blockIdx.x * blockDim.x + threadIdx.x;
    if (i >= n) return;
    int c = (int)(i % C);
    float mu  = sum[c] * invN;
    float var = sumsq[c] * invN - mu * mu;
    h[i] = (h[i] - mu) * rsqrtf(var + EPS_BN) * g[c] + b[c];
}

extern "C" void kernel_launch(void* const* d_in, const int* in_sizes, int n_in,
                              void* d_out, int out_size, void* d_ws, size_t ws_size,
                              hipStream_t stream) {
    const float* x    = (const float*)d_in[0];
    const int*   ei   = (const int*)  d_in[1];
    const float* Wg   = (const float*)d_in[2];
    const float* att  = (const float*)d_in[3];
    const float* bgat = (const float*)d_in[4];
    const float* g1   = (const float*)d_in[5];
    const float* b1   = (const float*)d_in[6];
    const float* g2   = (const float*)d_in[7];
    const float* b2   = (const float*)d_in[8];
    const float* g3   = (const float*)d_in[9];
    const float* b3   = (const float*)d_in[10];
    const float* W1   = (const float*)d_in[11];
    const float* W2   = (const float*)d_in[12];
    const float* W3   = (const float*)d_in[13];
    float* out = (float*)d_out;

    int E = in_sizes[1] / 2;
    int total = E + NNODES;

    float*    ws    = (float*)d_ws;
    float*    aggr  = ws + OFF_AGGR;             // GAT output accumulator -> h1
    unsigned* mmax  = (unsigned*)(ws + OFF_M);
    float*    denom = ws + OFF_DEN;
    float*    s1 = ws + OFF_SUM,       *q1 = s1 + F1;
    float*    s2 = q1 + F1,            *q2 = s2 + F1;
    float*    s3 = q2 + F1,            *q3 = s3 + F2;
    float*    h     = ws + OFF_H;                // h = x@Wg, later reused as h2
    float*    h3    = ws + OFF_H3;
    float*    ev    = ws + OFF_E;

    const float invN = 1.0f / (float)NNODES;
    const int waves80 = (NNODES / 16) * 5;       // N=80 -> 5 col tiles
    const int waves40 = (NNODES / 16) * 3;       // N=40 -> 3 col tiles (guarded)
    const int waves32 = (NNODES / 16) * 2;       // N=32 -> 2 col tiles

    // 0) zero accumulators (aggr, mmax encoded min == 0, denom, BN sums)
    zero_kernel<<<(int)((ZEROLEN + 255) / 256), 256, 0, stream>>>(ws, ZEROLEN);

    // 1) h = x @ W_gat            (50000x128 @ 128x80)
    gemm_wmma<<<(waves80 + 3) / 4, 128, 0, stream>>>(x, Wg, h, NNODES, OBS, F1, 5, 0);

    // 2) edge scores + segment max
    edge_score_kernel<<<(total + 255) / 256, 256, 0, stream>>>(h, ei, att, ev, mmax, E, total);

    // 3) exp + segment sum
    edge_exp_kernel<<<(total + 255) / 256, 256, 0, stream>>>(ev, ei, mmax, denom, E, total);

    // 4) weighted aggregation (wave per edge)
    edge_aggr_kernel<<<(total + 7) / 8, 256, 0, stream>>>(h, ei, ev, denom, aggr, E, total);

    // 5) + bias_gat -> h1 (in aggr)
    bias_add_kernel<<<(NNODES * F1 + 255) / 256, 256, 0, stream>>>(aggr, bgat, (long)NNODES * F1, F1);

    // 6) BN1 stats + apply, then h2 = relu(h1bn @ W1) into h buffer
    colstats_kernel<<<(NNODES + 127) / 128, F1, 0, stream>>>(aggr, NNODES, F1, s1, q1);
    bn_kernel<<<(NNODES * F1 + 255) / 256, 256, 0, stream>>>(aggr, g1, b1, s1, q1, (long)NNODES * F1, F1, invN);
    gemm_wmma<<<(waves80 + 3) / 4, 128, 0, stream>>>(aggr, W1, h, NNODES, F1, F1, 5, 1);

    // 7) BN2 + h3 = relu(h2bn @ W2)
    colstats_kernel<<<(NNODES + 127) / 128, F1, 0, stream>>>(h, NNODES, F1, s2, q2);
    bn_kernel<<<(NNODES * F1 + 255) / 256, 256, 0, stream>>>(h, g2, b2, s2, q2, (long)NNODES * F1, F1, invN);
    gemm_wmma<<<(waves40 + 3) / 4, 128, 0, stream>>>(h, W2, h3, NNODES, F1, F2, 3, 1);

    // 8) BN3 + out = sigmoid(relu(h3bn @ W3))
    colstats_kernel<<<(NNODES + 127) / 128, F2, 0, stream>>>(h3, NNODES, F2, s3, q3);
    bn_kernel<<<(NNODES * F2 + 255) / 256, 256, 0, stream>>>(h3, g3, b3, s3, q3, (long)NNODES * F2, F2, invN);
    gemm_wmma<<<(waves32 + 3) / 4, 128, 0, stream>>>(h3, W3, out, NNODES, F2, AC, 2, 2);
}